// W2VBert2Processor_5669356830877
// MI455X (gfx1250) — compile-verified
//
#include <hip/hip_runtime.h>
#include <hip/hip_bf16.h>
#include <math.h>

typedef __attribute__((ext_vector_type(2))) float v2f;
typedef __attribute__((ext_vector_type(4))) float v4f;
typedef __attribute__((ext_vector_type(8))) float v8f;

#define B_SZ 16
#define T_SZ 480000
#define FRAME_LEN 400
#define HOP 160
#define FFT_LEN 512
#define NBINS 257
#define NPAD 272              // DFT matrix N padded to 17 tiles of 16
#define NMEL 80
#define F_FRAMES 2998         // 1 + (480000-400)/160
#define NTOT 2998             // P = 0
#define HALF_T (NTOT/2)       // 1499
#define MEL_FLOOR 1.192092955078125e-07f
#define PREEMPH 0.97f

// ---- packed table layouts -------------------------------------------------
// DFT packed: v4f element  [(g*2 + h)*NPAD + n] = { cos(k,n), cos(k+1,n),
//                                                   sin(k,n), sin(k+1,n) }
// where g = k-group (k0 = 4g), h = lane-half (kOff = 2h), k = 4g + 2h.
// One lane's whole Re+Im fragment for a k-step = one 16B load.
#define NGRP_DFT (FFT_LEN/4)                    // 128
#define NV4_DFT  (NGRP_DFT*2*NPAD)              // 69632 v4f = 278528 floats
// MEL packed: v2f element [(g*2 + h)*NMEL + n] = { mel[k][n], mel[k+1][n] }
#define NGRP_MEL 68                             // covers K rows 0..271
#define NV2_MEL  (NGRP_MEL*2*NMEL)              // 10880 v2f = 21760 floats
#define NSTG_MEL 65                             // K stages used: k0 = 0..256

// workspace layout (floats)
#define OFF_DFTP 0
#define OFF_MELP (NV4_DFT*4)                    // 278528
#define OFF_MOUT (OFF_MELP + NV2_MEL*2)         // 300288
// mel_out: B*F*NMEL = 3,837,440 floats -> total ~16.6 MB

// ---------------------------------------------------------------------------
// Kernel 1: build packed DFT cos/sin table and packed mel weights in ws.
// Angle reduced mod 512 so cosf/sinf see args < 2*pi (full fp32 accuracy).
// ---------------------------------------------------------------------------
__global__ __launch_bounds__(256) void init_tables(
    const float* __restrict__ mel_filters, float* __restrict__ ws) {
  int i = blockIdx.x * blockDim.x + threadIdx.x;
  if (i < NV4_DFT) {
    int g = i / (2 * NPAD);
    int r = i % (2 * NPAD);
    int h = r / NPAD;
    int n = r % NPAD;
    int k = 4 * g + 2 * h;
    v4f val = {0.0f, 0.0f, 0.0f, 0.0f};
    if (n < NBINS) {
      int m0 = (k * n) & (FFT_LEN - 1);
      int m1 = ((k + 1) * n) & (FFT_LEN - 1);
      float a0 = (float)m0 * (6.28318530717958647692f / (float)FFT_LEN);
      float a1 = (float)m1 * (6.28318530717958647692f / (float)FFT_LEN);
      val.x = cosf(a0);
      val.y = cosf(a1);
      val.z = sinf(a0);
      val.w = sinf(a1);
    }
    ((v4f*)(ws + OFF_DFTP))[i] = val;
  } else {
    int j = i - NV4_DFT;
    if (j < NV2_MEL) {
      int g = j / (2 * NMEL);
      int r = j % (2 * NMEL);
      int h = r / NMEL;
      int n = r % NMEL;
      int k = 4 * g + 2 * h;
      v2f val;
      val.x = (k < NBINS) ? mel_filters[k * NMEL + n] : 0.0f;
      val.y = (k + 1 < NBINS) ? mel_filters[(k + 1) * NMEL + n] : 0.0f;
      ((v2f*)(ws + OFF_MELP))[j] = val;
    }
  }
}

// ---------------------------------------------------------------------------
// Kernel 2: 16 frames per workgroup (8 waves).
//  Phase 1: preprocess frames into LDS (scale, de-mean, preemph, window, pad)
//  Phase 2: DFT as fp32 WMMA GEMM (depth-2 pipeline, b128 B loads)
//  Phase 3: mel projection as fp32 WMMA GEMM -> log-mel to workspace
// ---------------------------------------------------------------------------
__global__ __launch_bounds__(256) void frames_to_mel(
    const float* __restrict__ wav, const float* __restrict__ window,
    float* __restrict__ ws) {
  // pitch 516 words: lane stride 516 % 64 == 4 -> 16 distinct LDS banks
  __shared__ float ldsFr[16][516];
  // dual use: phase 1 = raw frame samples (400 cols); phase 2/3 = power
  // spectrum (272 cols). pitch 404 % 64 == 20 -> conflict-free column reads.
  __shared__ float ldsXS[16][404];

  float* __restrict__ melOut = ws + OFF_MOUT;

  const int b   = blockIdx.y;
  const int f0  = blockIdx.x * 16;
  const int tid = threadIdx.x;
  const int wave = tid >> 5;
  const int lane = tid & 31;

  // ---- Phase 1: preprocess (each wave owns 2 frames) ----
  for (int fi = wave; fi < 16; fi += 8) {
    const int f = f0 + fi;
    const bool valid = f < F_FRAMES;
    const float* src = wav + (size_t)b * T_SZ + (size_t)f * HOP;
    float sum = 0.0f;
    for (int i = lane; i < FRAME_LEN; i += 32) {
      float v = valid ? src[i] * 32768.0f : 0.0f;
      ldsXS[fi][i] = v;
      sum += v;
    }
    // wave32 reduction for the frame mean
    for (int off = 16; off > 0; off >>= 1) sum += __shfl_down(sum, off, 32);
    const float mean = __shfl(sum, 0, 32) * (1.0f / (float)FRAME_LEN);
    for (int i = lane; i < FFT_LEN; i += 32) {
      float fr = 0.0f;
      if (i < FRAME_LEN) {
        float x = ldsXS[fi][i] - mean;
        float pre = (i == 0) ? x * (1.0f - PREEMPH)
                             : x - PREEMPH * (ldsXS[fi][i - 1] - mean);
        fr = pre * window[i];
      }
      ldsFr[fi][i] = fr;
    }
  }
  __syncthreads();

  // fp32 WMMA 16x16x4 lane mapping
  const int laneM = lane & 15;        // M row (A) / N col (B,D)
  const int half  = lane >> 4;        // 0: K+0,K+1   1: K+2,K+3
  const int kOff  = half * 2;
  const int rowBase = half * 8;

  // ---- Phase 2: DFT GEMM, Re & Im accumulated together, 17 N-tiles ----
  for (int nt = wave; nt < NPAD / 16; nt += 8) {
    const int n0 = nt * 16;
    const v4f* __restrict__ bP =
        (const v4f*)(ws + OFF_DFTP) + (size_t)half * NPAD + (n0 + laneM);
    __builtin_prefetch(bP, 0, 2);  // warm L2 path for this tile's column
    v8f accR = {};
    v8f accI = {};
    // depth-2 software pipeline: stage g+2 loads issue before stage g is
    // consumed -> two distinct in-flight B buffers, no full loadcnt drains.
    v2f a0, a1;
    a0.x = ldsFr[laneM][kOff];
    a0.y = ldsFr[laneM][kOff + 1];
    a1.x = ldsFr[laneM][4 + kOff];
    a1.y = ldsFr[laneM][4 + kOff + 1];
    v4f b0 = bP[0];
    v4f b1 = bP[(size_t)2 * NPAD];
#pragma unroll 4
    for (int g = 0; g < NGRP_DFT - 2; ++g) {
      const int k2 = (g + 2) * 4;
      v2f a2;
      a2.x = ldsFr[laneM][k2 + kOff];
      a2.y = ldsFr[laneM][k2 + kOff + 1];
      v4f b2 = bP[(size_t)(g + 2) * 2 * NPAD];
      v2f bR; bR.x = b0.x; bR.y = b0.y;
      v2f bI; bI.x = b0.z; bI.y = b0.w;
      accR = __builtin_amdgcn_wmma_f32_16x16x4_f32(
          false, a0, false, bR, (short)0, accR, false, false);
      accI = __builtin_amdgcn_wmma_f32_16x16x4_f32(
          false, a0, false, bI, (short)0, accI, false, false);
      a0 = a1; a1 = a2;
      b0 = b1; b1 = b2;
    }
    {  // peeled epilogue: two remaining stages
      v2f bR; bR.x = b0.x; bR.y = b0.y;
      v2f bI; bI.x = b0.z; bI.y = b0.w;
      accR = __builtin_amdgcn_wmma_f32_16x16x4_f32(
          false, a0, false, bR, (short)0, accR, false, false);
      accI = __builtin_amdgcn_wmma_f32_16x16x4_f32(
          false, a0, false, bI, (short)0, accI, false, false);
      bR.x = b1.x; bR.y = b1.y;
      bI.x = b1.z; bI.y = b1.w;
      accR = __builtin_amdgcn_wmma_f32_16x16x4_f32(
          false, a1, false, bR, (short)0, accR, false, false);
      accI = __builtin_amdgcn_wmma_f32_16x16x4_f32(
          false, a1, false, bI, (short)0, accI, false, false);
    }
    // |X|^2 into spec tile (ldsXS reused; cols >= 257 are exactly 0)
    for (int v = 0; v < 8; ++v) {
      float re = accR[v], im = accI[v];
      ldsXS[rowBase + v][n0 + laneM] = re * re + im * im;
    }
  }
  __syncthreads();

  // ---- Phase 3: mel GEMM (16x260 @ 260x80), 5 N-tiles, then log ----
  for (int nt = wave; nt < NMEL / 16; nt += 8) {
    const int n0 = nt * 16;
    const v2f* __restrict__ mP =
        (const v2f*)(ws + OFF_MELP) + (size_t)half * NMEL + (n0 + laneM);
    v8f acc = {};
    v2f a0, a1;
    a0.x = ldsXS[laneM][kOff];
    a0.y = ldsXS[laneM][kOff + 1];
    a1.x = ldsXS[laneM][4 + kOff];
    a1.y = ldsXS[laneM][4 + kOff + 1];
    v2f m0 = mP[0];
    v2f m1 = mP[(size_t)2 * NMEL];
#pragma unroll 4
    for (int g = 0; g < NSTG_MEL - 2; ++g) {  // K rows >= 257 are zero-padded
      const int k2 = (g + 2) * 4;
      v2f a2;
      a2.x = ldsXS[laneM][k2 + kOff];
      a2.y = ldsXS[laneM][k2 + kOff + 1];
      v2f m2 = mP[(size_t)(g + 2) * 2 * NMEL];
      acc = __builtin_amdgcn_wmma_f32_16x16x4_f32(
          false, a0, false, m0, (short)0, acc, false, false);
      a0 = a1; a1 = a2;
      m0 = m1; m1 = m2;
    }
    acc = __builtin_amdgcn_wmma_f32_16x16x4_f32(
        false, a0, false, m0, (short)0, acc, false, false);
    acc = __builtin_amdgcn_wmma_f32_16x16x4_f32(
        false, a1, false, m1, (short)0, acc, false, false);
    for (int v = 0; v < 8; ++v) {
      const int f = f0 + rowBase + v;
      if (f < F_FRAMES) {
        float m = fmaxf(acc[v], MEL_FLOOR);
        melOut[((size_t)b * F_FRAMES + f) * NMEL + n0 + laneM] = logf(m);
      }
    }
  }
}

// ---------------------------------------------------------------------------
// Kernel 3: per-(batch, mel) mean/var (ddof=1) normalization + mask.
// Reshape (B, 2998, 80) -> (B, 1499, 160) is contiguous, so writes are flat.
// ---------------------------------------------------------------------------
__global__ __launch_bounds__(256) void normalize_write(
    const float* __restrict__ ws, float* __restrict__ out) {
  const float* __restrict__ melOut = ws + OFF_MOUT;
  const int b = blockIdx.x / NMEL;
  const int m = blockIdx.x % NMEL;
  const int tid = threadIdx.x;
  __shared__ float rs[256];
  __shared__ float rs2[256];

  float s = 0.0f, s2 = 0.0f;
  for (int f = tid; f < F_FRAMES; f += 256) {
    float v = melOut[((size_t)b * F_FRAMES + f) * NMEL + m];
    s += v;
    s2 += v * v;
  }
  rs[tid] = s;
  rs2[tid] = s2;
  __syncthreads();
  for (int st = 128; st > 0; st >>= 1) {
    if (tid < st) { rs[tid] += rs[tid + st]; rs2[tid] += rs2[tid + st]; }
    __syncthreads();
  }
  const float mean = rs[0] * (1.0f / (float)F_FRAMES);
  const float var = (rs2[0] - mean * rs[0]) / (float)(F_FRAMES - 1);
  const float inv = rsqrtf(var + 1e-7f);

  for (int f = tid; f < F_FRAMES; f += 256) {
    float v = melOut[((size_t)b * F_FRAMES + f) * NMEL + m];
    out[((size_t)b * NTOT + f) * NMEL + m] = (v - mean) * inv;
  }
  // attention mask (int32 ones; F == Ntot so no padded frames)
  if (m == 0) {
    int* mask = (int*)(out + (size_t)B_SZ * NTOT * NMEL);
    for (int j = tid; j < HALF_T; j += 256) mask[(size_t)b * HALF_T + j] = 1;
  }
}

extern "C" void kernel_launch(void* const* d_in, const int* in_sizes, int n_in,
                              void* d_out, int out_size, void* d_ws,
                              size_t ws_size, hipStream_t stream) {
  const float* raw_speech  = (const float*)d_in[0];
  const float* mel_filters = (const float*)d_in[1];
  const float* window      = (const float*)d_in[2];
  float* ws  = (float*)d_ws;
  float* out = (float*)d_out;

  const int totalInit = NV4_DFT + NV2_MEL;
  hipLaunchKernelGGL(init_tables, dim3((totalInit + 255) / 256), dim3(256), 0,
                     stream, mel_filters, ws);

  dim3 gridB((F_FRAMES + 15) / 16, B_SZ);
  hipLaunchKernelGGL(frames_to_mel, gridB, dim3(256), 0, stream, raw_speech,
                     window, ws);

  hipLaunchKernelGGL(normalize_write, dim3(B_SZ * NMEL), dim3(256), 0, stream,
                     ws, out);
}